// VQC_21182778704492
// MI455X (gfx1250) — compile-verified
//
#include <hip/hip_runtime.h>

// VQC simulator for MI455X (gfx1250).
// One workgroup per batch sample; full 2^16 f32 state vector lives in LDS
// (256KB of the WGP's 320KB). All 16-qubit RY layers + within-group CNOTs are
// fused into four 16x16 group matrices applied along 4-qubit axes with
// V_WMMA_F32_16X16X4_F32. Cross-group CNOTs are XOR-mask swap passes.

typedef __attribute__((ext_vector_type(2))) float v2f;
typedef __attribute__((ext_vector_type(8))) float v8f;

#define NQ      16
#define NSTATE  65536      // 2^16 amplitudes
#define BLOCK   512        // 16 waves of 32
#define NWAVES  16
#define NCLASS  10

// Within-group CNOT permutation on a nibble v = (b3 b2 b1 b0),
// qubit 4g+j <-> nibble bit (3-j).
// Sequence: CNOT(4g,4g+1), CNOT(4g+2,4g+3) [even layer], CNOT(4g+1,4g+2) [odd].
__device__ __forceinline__ int perm4(int v) {
  v ^= ((v >> 3) & 1) << 2;  // ctrl b3 -> flip b2
  v ^= ((v >> 1) & 1) << 0;  // ctrl b1 -> flip b0
  v ^= ((v >> 2) & 1) << 1;  // ctrl b2 -> flip b1 (after even layer)
  return v;
}

__global__ __launch_bounds__(BLOCK)
void vqc_wmma_kernel(const float* __restrict__ x,
                     const float* __restrict__ w,
                     float* __restrict__ out) {
  extern __shared__ float lds[];
  float* state = lds;               // 65536 floats
  float* mats  = lds + NSTATE;      // 4 * 256 floats (group matrices)
  float* csc   = mats + 1024;       // 16 cos(theta/2)
  float* css   = csc + 16;          // 16 sin(theta/2)
  float* red   = css + 16;          // 10 reduction slots

  const int tid  = threadIdx.x;
  const int b    = blockIdx.x;
  const int lane = tid & 31;
  const int wav  = tid >> 5;

  // H^{x16}|0..0> : uniform real amplitude 2^-8
  for (int i = tid; i < NSTATE; i += BLOCK) state[i] = 0.00390625f;
  __syncthreads();

  // 9 matmul stages: s=0 -> P*RY4(x); s=1..7 -> P*RY4(w[s-1]); s=8 -> RY4(w[7]).
  // Cross-group CNOT swap pass after stages 0..7.
  for (int s = 0; s < 9; ++s) {
    if (tid < NQ) {
      float th = (s == 0) ? x[b * NQ + tid] : w[(s - 1) * NQ + tid];
      float h = 0.5f * th;
      csc[tid] = __cosf(h);
      css[tid] = __sinf(h);
    }
    __syncthreads();

    // Build the four 16x16 group matrices: RY4 tensor product, rows scattered
    // through the within-group CNOT permutation (except the last stage).
    for (int e = tid; e < 1024; e += BLOCK) {
      int g = e >> 8;
      int rc = e & 255;
      int r = rc >> 4, c = rc & 15;
      float val = 1.0f;
#pragma unroll
      for (int j = 0; j < 4; ++j) {
        int q = 4 * g + j;
        int rj = (r >> (3 - j)) & 1;
        int cj = (c >> (3 - j)) & 1;
        float cv = csc[q], sv = css[q];
        // RY = [[c,-s],[s,c]]
        val *= (rj == cj) ? cv : ((rj == 0) ? -sv : sv);
      }
      int rp = (s < 8) ? perm4(r) : r;
      mats[g * 256 + rp * 16 + c] = val;
    }
    __syncthreads();

    // Apply each group matrix along its 4-qubit axis via WMMA f32 16x16x4.
    const int m    = lane & 15;           // A row / output col-lane
    const int koff = (lane >> 4) << 1;    // lanes 0-15: K 0,1 ; lanes 16-31: K 2,3
    for (int g = 0; g < 4; ++g) {
      const int sh = 4 * (3 - g);         // bit shift of this group's nibble
      const float* Mg = mats + g * 256;
      // A-matrix fragments for the 4 K-chunks (constant over tiles)
      v2f a0, a1, a2, a3;
      a0[0] = Mg[m * 16 + 0  + koff]; a0[1] = Mg[m * 16 + 1  + koff];
      a1[0] = Mg[m * 16 + 4  + koff]; a1[1] = Mg[m * 16 + 5  + koff];
      a2[0] = Mg[m * 16 + 8  + koff]; a2[1] = Mg[m * 16 + 9  + koff];
      a3[0] = Mg[m * 16 + 12 + koff]; a3[1] = Mg[m * 16 + 13 + koff];

      for (int t = wav; t < 256; t += NWAVES) {      // 4096/16 column tiles
        int rest = t * 16 + m;                       // this lane's column
        int lo   = rest & ((1 << sh) - 1);
        int hi   = rest >> sh;
        int base = (hi << (sh + 4)) | lo;            // nibble slot zeroed
        // B fragments: S[k][n] at address base + (k << sh)
        v2f b0, b1, b2, b3;
        b0[0] = state[base + ((0  + koff) << sh)];
        b0[1] = state[base + ((1  + koff) << sh)];
        b1[0] = state[base + ((4  + koff) << sh)];
        b1[1] = state[base + ((5  + koff) << sh)];
        b2[0] = state[base + ((8  + koff) << sh)];
        b2[1] = state[base + ((9  + koff) << sh)];
        b3[0] = state[base + ((12 + koff) << sh)];
        b3[1] = state[base + ((13 + koff) << sh)];
        v8f d = {};
        d = __builtin_amdgcn_wmma_f32_16x16x4_f32(false, a0, false, b0, (short)0, d, false, false);
        d = __builtin_amdgcn_wmma_f32_16x16x4_f32(false, a1, false, b1, (short)0, d, false, false);
        d = __builtin_amdgcn_wmma_f32_16x16x4_f32(false, a2, false, b2, (short)0, d, false, false);
        d = __builtin_amdgcn_wmma_f32_16x16x4_f32(false, a3, false, b3, (short)0, d, false, false);
        // C/D layout: VGPR i -> row i (lanes 0-15) or i+8 (lanes 16-31)
        const int mbase = (lane >> 4) << 3;
#pragma unroll
        for (int i = 0; i < 8; ++i)
          state[base + ((mbase + i) << sh)] = d[i];
      }
      __syncthreads();
    }

    if (s < 8) {
      // Cross-group CNOTs (3,4),(7,8),(11,12): qubit q <-> bit (15-q).
      // Involution: partner = idx ^ mask(ctrl bits 12,8,4 -> tgt bits 11,7,3).
      for (int i = tid; i < NSTATE; i += BLOCK) {
        int mask = (((i >> 12) & 1) << 11) | (((i >> 8) & 1) << 7) | (((i >> 4) & 1) << 3);
        int p = i ^ mask;
        if (p > i) {
          float t0 = state[i], t1 = state[p];
          state[i] = t1;
          state[p] = t0;
        }
      }
      __syncthreads();
    }
  }

  // <Z_q> for q = 0..9 : sum of (+/-) state^2 by bit (15-q)
  if (tid < NCLASS) red[tid] = 0.0f;
  __syncthreads();
  float acc[NCLASS];
#pragma unroll
  for (int q = 0; q < NCLASS; ++q) acc[q] = 0.0f;
  for (int i = tid; i < NSTATE; i += BLOCK) {
    float v = state[i];
    float pr = v * v;
#pragma unroll
    for (int q = 0; q < NCLASS; ++q)
      acc[q] += ((i >> (15 - q)) & 1) ? -pr : pr;
  }
#pragma unroll
  for (int q = 0; q < NCLASS; ++q) {
    float v = acc[q];
    v += __shfl_xor(v, 16);
    v += __shfl_xor(v, 8);
    v += __shfl_xor(v, 4);
    v += __shfl_xor(v, 2);
    v += __shfl_xor(v, 1);
    if (lane == 0) atomicAdd(&red[q], v);
  }
  __syncthreads();
  if (tid < NCLASS) out[b * NCLASS + tid] = red[tid];
}

extern "C" void kernel_launch(void* const* d_in, const int* in_sizes, int n_in,
                              void* d_out, int out_size, void* d_ws, size_t ws_size,
                              hipStream_t stream) {
  const float* x = (const float*)d_in[0];   // (1024, 16) f32
  const float* w = (const float*)d_in[1];   // (8, 16) f32
  float* out = (float*)d_out;               // (1024, 10) f32
  int B = in_sizes[0] / NQ;
  size_t shmem = (size_t)(NSTATE + 1024 + 16 + 16 + NCLASS) * sizeof(float);
  vqc_wmma_kernel<<<B, BLOCK, shmem, stream>>>(x, w, out);
  (void)n_in; (void)out_size; (void)d_ws; (void)ws_size;
}